// KANLSTMCell_89953795047907
// MI455X (gfx1250) — compile-verified
//
#include <hip/hip_runtime.h>
#include <hip/hip_bf16.h>

typedef float v2f __attribute__((ext_vector_type(2)));
typedef float v8f __attribute__((ext_vector_type(8)));

#define B_  2048
#define I_  256
#define H_  512
#define K_  32
#define D_  768
#define DC  64                 // D-chunk (WMMA K panel held in LDS)
#define NCHUNK (D_ / DC)       // 12
#define LDSPAD 17              // stride pad: conflict-free A reads

__device__ __forceinline__ float sigmoidf_(float v) {
    return 1.0f / (1.0f + __expf(-v));
}

__global__ __launch_bounds__(256, 1)
void kan_lstm_fused_kernel(const float* __restrict__ x,
                           const float* __restrict__ h_prev,
                           const float* __restrict__ c_prev,
                           const float* __restrict__ W1,
                           const float* __restrict__ b1,
                           const float* __restrict__ W2,
                           const float* __restrict__ b2,
                           const float* __restrict__ Wc,
                           const float* __restrict__ bc,
                           float* __restrict__ out) {
    // U[g][d_local][row], padded to kill LDS bank conflicts: 4*64*17*4B = 17.4KB
    __shared__ float U[4 * DC * LDSPAD];

    const int tid  = threadIdx.x;
    const int lane = tid & 31;
    const int wave = tid >> 5;           // 0..7
    const int row0 = blockIdx.x * 16;    // 16 batch rows per block

    // ---- stage-1 mapping: thread <-> (gate, d_local) ----
    const int g1 = tid >> 6;             // 0..3
    const int dl = tid & 63;             // 0..63

    // ---- stage-2 mapping: wave <-> 64-wide column strip ----
    const int n0    = wave * 64;
    const int m     = lane & 15;               // M (A) / N (B,C) sub-index
    const int khalf = (lane >> 4) << 1;        // 0 or 2 (K offset per lane half)

    v8f acc[4][4] = {};                        // 4 gates x 4 col-tiles, f32 accum

    for (int c = 0; c < NCHUNK; ++c) {
        // ================= stage 1: per-feature MLP -> U (LDS) =================
        {
            const int d = c * DC + dl;                       // 0..767, chunk-uniform side
            const float* cb;
            int stride;
            if (d < I_) { cb = x      + (size_t)row0 * I_ + d;        stride = I_; }
            else        { cb = h_prev + (size_t)row0 * H_ + (d - I_); stride = H_; }

            float cr[16];
            #pragma unroll
            for (int r = 0; r < 16; ++r) cr[r] = cb[(size_t)r * stride];

            float u[16];
            const float bb2 = b2[g1 * D_ + d];
            #pragma unroll
            for (int r = 0; r < 16; ++r) u[r] = bb2;

            const int wbase = (g1 * D_ + d) * K_;
            #pragma unroll 4
            for (int k = 0; k < K_; ++k) {
                const float w1 = W1[wbase + k];
                const float bb = b1[wbase + k];
                const float w2 = W2[wbase + k];
                #pragma unroll
                for (int r = 0; r < 16; ++r) {
                    const float hr = fmaxf(fmaf(cr[r], w1, bb), 0.0f);
                    u[r] = fmaf(hr, w2, u[r]);
                }
            }

            __syncthreads();   // previous chunk's stage-2 reads done
            const int ub = (g1 * DC + dl) * LDSPAD;
            #pragma unroll
            for (int r = 0; r < 16; ++r) U[ub + r] = u[r];
        }
        __syncthreads();       // U chunk visible to all waves

        // ================= stage 2: f32 WMMA over this K=64 panel =================
        #pragma unroll
        for (int g = 0; g < 4; ++g) {
            const float* wcg = Wc + ((size_t)g * D_ + (size_t)c * DC) * H_;
            #pragma unroll 4
            for (int s = 0; s < 16; ++s) {            // 16 steps of K=4
                const int kk = 4 * s + khalf;
                // A fragment (16x4 f32): lanes0-15 M=lane,K=4s+{0,1}; lanes16-31 K=4s+{2,3}
                v2f a;
                a.x = U[(g * DC + kk    ) * LDSPAD + m];
                a.y = U[(g * DC + kk + 1) * LDSPAD + m];
                #pragma unroll
                for (int t = 0; t < 4; ++t) {
                    const int ncol = n0 + t * 16 + m;
                    // B fragment (4x16 f32): lanes hold N=m, K rows kk/kk+1
                    v2f bf;
                    bf.x = wcg[(size_t)(kk    ) * H_ + ncol];
                    bf.y = wcg[(size_t)(kk + 1) * H_ + ncol];
                    acc[g][t] = __builtin_amdgcn_wmma_f32_16x16x4_f32(
                        false, a, false, bf, (short)0, acc[g][t], false, false);
                }
            }
        }
    }

    // ================= LSTM combine: all 4 gates in-register =================
    #pragma unroll
    for (int t = 0; t < 4; ++t) {
        const int col = n0 + t * 16 + m;
        const float bcf = bc[0 * H_ + col];
        const float bci = bc[1 * H_ + col];
        const float bco = bc[2 * H_ + col];
        const float bcc = bc[3 * H_ + col];
        #pragma unroll
        for (int j = 0; j < 8; ++j) {
            const int row = j + ((lane >> 4) << 3);          // C/D layout: M = j (+8)
            const size_t idx = (size_t)(row0 + row) * H_ + col;
            const float fg = sigmoidf_(acc[0][t][j] + bcf);
            const float ig = sigmoidf_(acc[1][t][j] + bci);
            const float og = sigmoidf_(acc[2][t][j] + bco);
            const float cg = tanhf   (acc[3][t][j] + bcc);
            const float cp = c_prev[idx];
            const float cn = fg * cp + ig * cg;
            out[idx] = og * tanhf(cn);                       // h_t
            out[(size_t)B_ * H_ + idx] = cn;                 // c_t
        }
    }
}

extern "C" void kernel_launch(void* const* d_in, const int* in_sizes, int n_in,
                              void* d_out, int out_size, void* d_ws, size_t ws_size,
                              hipStream_t stream) {
    const float* x      = (const float*)d_in[0];
    const float* h_prev = (const float*)d_in[1];
    const float* c_prev = (const float*)d_in[2];
    const float* W1     = (const float*)d_in[3];
    const float* b1     = (const float*)d_in[4];
    const float* W2     = (const float*)d_in[5];
    const float* b2     = (const float*)d_in[6];
    const float* Wc     = (const float*)d_in[7];
    const float* bc     = (const float*)d_in[8];
    float* out = (float*)d_out;

    dim3 grid(B_ / 16);   // 128 blocks, each owns 16 batch rows
    dim3 block(256);      // 8 waves: each wave owns a 64-wide strip of H
    kan_lstm_fused_kernel<<<grid, block, 0, stream>>>(
        x, h_prev, c_prev, W1, b1, W2, b2, Wc, bc, out);
}